// ExCL_22771916604009
// MI455X (gfx1250) — compile-verified
//
#include <hip/hip_runtime.h>
#include <hip/hip_bf16.h>
#include <math.h>

typedef __attribute__((ext_vector_type(2))) float v2f;
typedef __attribute__((ext_vector_type(8))) float v8f;

#define WMMA_F32(a, b, c) \
  __builtin_amdgcn_wmma_f32_16x16x4_f32(false, (a), false, (b), (short)0, (c), false, false)

__device__ __forceinline__ float sigm(float x) { return 1.0f / (1.0f + expf(-x)); }

// ---------------------------------------------------------------------------
// Generic f32 WMMA GEMM: C[M,N] = A[M,K] @ W[N,K]^T (+bias / fused head epilogue)
// block = 256 threads (8 waves), macro-tile 64(M) x 128(N), K-tile 32.
// MODE 0: Cout[m*N+n] = acc + bias[n]
// MODE 2: h = tanh(acc + P[(m>>8)*256 + n]); atomicAdd(logits[m], sum_n h*w2[n])
// ---------------------------------------------------------------------------
template <int MODE>
__global__ __launch_bounds__(256) void gemm_wmma(
    const float* __restrict__ A, const float* __restrict__ W,
    const float* __restrict__ bias, float* __restrict__ Cout,
    const float* __restrict__ Pmat, const float* __restrict__ w2,
    float* __restrict__ logits, int M, int N, int K) {
  __shared__ float As[32][65];
  __shared__ float Bs[32][129];
  const int tid = threadIdx.x;
  const int lane = tid & 31, wid = tid >> 5;
  const int lm = lane & 15, half = lane >> 4;
  const int wy = wid >> 1, wx = wid & 1;
  const int m0 = blockIdx.y * 64, n0 = blockIdx.x * 128;

  v8f acc[4];
#pragma unroll
  for (int s = 0; s < 4; ++s) acc[s] = (v8f){0.f, 0.f, 0.f, 0.f, 0.f, 0.f, 0.f, 0.f};

  const int j = tid & 31;
  const int i0 = tid >> 5;
  for (int kb = 0; kb < K; kb += 32) {
    const int kk = kb + j;
    const bool kin = (kk < K);
#pragma unroll
    for (int it = 0; it < 8; ++it) {
      int i = i0 + it * 8;
      As[j][i] = kin ? A[(size_t)(m0 + i) * K + kk] : 0.f;
    }
#pragma unroll
    for (int it = 0; it < 16; ++it) {
      int nn = i0 + it * 8;
      Bs[j][nn] = kin ? W[(size_t)(n0 + nn) * K + kk] : 0.f;
    }
    __syncthreads();
#pragma unroll
    for (int ki = 0; ki < 32; ki += 4) {
      const int k0 = ki + half * 2;
      v2f a;
      a.x = As[k0][wy * 16 + lm];
      a.y = As[k0 + 1][wy * 16 + lm];
#pragma unroll
      for (int s = 0; s < 4; ++s) {
        const int nc = wx * 64 + s * 16 + lm;
        v2f bb;
        bb.x = Bs[k0][nc];
        bb.y = Bs[k0 + 1][nc];
        acc[s] = WMMA_F32(a, bb, acc[s]);
      }
    }
    __syncthreads();
  }

  if (MODE == 0) {
#pragma unroll
    for (int s = 0; s < 4; ++s) {
      const int n = n0 + wx * 64 + s * 16 + lm;
      const float bv = bias[n];
#pragma unroll
      for (int r = 0; r < 8; ++r) {
        const int mr = m0 + wy * 16 + r + half * 8;
        Cout[(size_t)mr * N + n] = acc[s][r] + bv;
      }
    }
  } else {
#pragma unroll
    for (int r = 0; r < 8; ++r) {
      const int mr = m0 + wy * 16 + r + half * 8;
      const int bidx = mr >> 8;  // T == 256
      float part = 0.f;
#pragma unroll
      for (int s = 0; s < 4; ++s) {
        const int n = n0 + wx * 64 + s * 16 + lm;
        const float h = tanhf(acc[s][r] + Pmat[bidx * 256 + n]);
        part += h * w2[n];
      }
      part += __shfl_xor(part, 1, 32);
      part += __shfl_xor(part, 2, 32);
      part += __shfl_xor(part, 4, 32);
      part += __shfl_xor(part, 8, 32);
      if (lm == 0) atomicAdd(&logits[mr], part);
    }
  }
}

// ---------------------------------------------------------------------------
// LSTM recurrence. Batch split into 8 workgroups of 16 rows (independent).
// 512 threads = 16 waves; each wave owns one 16-hidden-column tile (4 gate
// accumulators). h slice in LDS, c in accumulator VGPRs. Gate pre-activations
// (x@Wih^T + bias) precomputed in gates_x. Wt2 is Whh^T packed so that rows
// (k, k+1) are interleaved: Wt2[(k>>1)*2048 + n*2 + (k&1)] -> B frag is one
// coalesced 64-bit load.
// mode 0: store h only at t == len-1 into out_last[B,256]  (lang)
// mode 1: store h * (t < len) every step into out_seq[B,T,256] (vid)
// ---------------------------------------------------------------------------
__global__ __launch_bounds__(512) void lstm_rec(
    const float* __restrict__ gates_x, const float* __restrict__ Wt2,
    const int* __restrict__ lengths, float* __restrict__ out_seq,
    float* __restrict__ out_last, int Tsteps, int mode) {
  __shared__ float hs[16][257];
  const int tid = threadIdx.x;
  const int lane = tid & 31, wid = tid >> 5;   // 16 waves
  const int lm = lane & 15, half = lane >> 4;
  const int b0 = blockIdx.x * 16;
  const int nc0 = wid * 16;                    // this wave's hidden-column tile
  const v2f* __restrict__ Wp = (const v2f*)Wt2;

  for (int i = tid; i < 16 * 257; i += 512) ((float*)hs)[i] = 0.f;
  v8f cst = (v8f){0.f, 0.f, 0.f, 0.f, 0.f, 0.f, 0.f, 0.f};
  int len[8];
#pragma unroll
  for (int r = 0; r < 8; ++r) len[r] = lengths[b0 + r + half * 8];
  __syncthreads();

  const int ncol = nc0 + lm;
  for (int t = 0; t < Tsteps; ++t) {
    v8f ac[4];
    // init accumulators with x-projection gates (C layout: m = r + 8*half)
#pragma unroll
    for (int r = 0; r < 8; ++r) {
      const size_t base = ((size_t)(b0 + r + half * 8) * Tsteps + t) * 1024;
#pragma unroll
      for (int g = 0; g < 4; ++g) ac[g][r] = gates_x[base + g * 256 + ncol];
    }
    // recurrent GEMM: += h_prev @ Whh^T
#pragma unroll 4
    for (int k = 0; k < 256; k += 4) {
      const int k0 = k + half * 2;             // even
      v2f a;
      a.x = hs[lm][k0];
      a.y = hs[lm][k0 + 1];
      const size_t prow = (size_t)(k0 >> 1) * 1024;  // v2f index of packed row-pair
#pragma unroll
      for (int g = 0; g < 4; ++g) {
        const v2f bb = Wp[prow + g * 256 + ncol];
        ac[g] = WMMA_F32(a, bb, ac[g]);
      }
    }
    // elementwise LSTM cell (PyTorch gate order i,f,g,o)
    float hnew[8];
#pragma unroll
    for (int r = 0; r < 8; ++r) {
      const float iv = sigm(ac[0][r]);
      const float fv = sigm(ac[1][r]);
      const float gv = tanhf(ac[2][r]);
      const float ov = sigm(ac[3][r]);
      const float cc = fv * cst[r] + iv * gv;
      cst[r] = cc;
      hnew[r] = ov * tanhf(cc);
    }
    __syncthreads();  // all reads of hs for step t done
#pragma unroll
    for (int r = 0; r < 8; ++r) {
      const int mrow = r + half * 8;
      hs[mrow][ncol] = hnew[r];
      if (mode == 1) {
        const float msk = (t < len[r]) ? 1.f : 0.f;
        out_seq[((size_t)(b0 + mrow) * Tsteps + t) * 256 + ncol] = hnew[r] * msk;
      } else {
        if (t == len[r] - 1)
          out_last[(size_t)(b0 + mrow) * 256 + ncol] = hnew[r];
      }
    }
    __syncthreads();
  }
}

// ---------------------------------------------------------------------------
// Small prep kernels
// ---------------------------------------------------------------------------
// Whh[1024][256] -> Wt2 packed: Wt2[(k>>1)*2048 + n*2 + (k&1)] = Whh[n][k]
__global__ void k_transpose_pack(const float* __restrict__ Whh, float* __restrict__ Wt2) {
  const int idx = blockIdx.x * 256 + threadIdx.x;  // 262144
  const int parity = idx & 1;
  const int n = (idx >> 1) & 1023;
  const int kpair = idx >> 11;
  const int k = kpair * 2 + parity;
  Wt2[idx] = Whh[n * 256 + k];
}
__global__ void k_add(const float* __restrict__ a, const float* __restrict__ b,
                      float* __restrict__ o, int n) {
  const int i = blockIdx.x * 256 + threadIdx.x;
  if (i < n) o[i] = a[i] + b[i];
}
__global__ void k_w1sum(const float* __restrict__ W1, float* __restrict__ Wa,
                        float* __restrict__ Wb) {
  const int idx = blockIdx.x * 256 + threadIdx.x;  // 65536
  const int n = idx >> 8, k = idx & 255;
  Wa[idx] = W1[n * 1024 + k] + W1[n * 1024 + 512 + k];        // W1a + W1c (lang part)
  Wb[idx] = W1[n * 1024 + 256 + k] + W1[n * 1024 + 768 + k];  // W1b + W1d (video part)
}

// ---------------------------------------------------------------------------
// Masked softmax + KL losses, one block per batch row (T == 256 threads)
// ---------------------------------------------------------------------------
__device__ __forceinline__ float block_sum(float v, float* sbuf) {
  const int lane = threadIdx.x & 31, w = threadIdx.x >> 5;
#pragma unroll
  for (int o = 16; o; o >>= 1) v += __shfl_xor(v, o, 32);
  if (lane == 0) sbuf[w] = v;
  __syncthreads();
  if (w == 0) {
    float y = (lane < 8) ? sbuf[lane] : 0.f;
#pragma unroll
    for (int o = 4; o; o >>= 1) y += __shfl_xor(y, o, 32);
    if (lane == 0) sbuf[0] = y;
  }
  __syncthreads();
  const float r = sbuf[0];
  __syncthreads();
  return r;
}
__device__ __forceinline__ float block_max(float v, float* sbuf) {
  const int lane = threadIdx.x & 31, w = threadIdx.x >> 5;
#pragma unroll
  for (int o = 16; o; o >>= 1) v = fmaxf(v, __shfl_xor(v, o, 32));
  if (lane == 0) sbuf[w] = v;
  __syncthreads();
  if (w == 0) {
    float y = (lane < 8) ? sbuf[lane] : -3.0e38f;
#pragma unroll
    for (int o = 4; o; o >>= 1) y = fmaxf(y, __shfl_xor(y, o, 32));
    if (lane == 0) sbuf[0] = y;
  }
  __syncthreads();
  const float r = sbuf[0];
  __syncthreads();
  return r;
}
__device__ __forceinline__ int block_argmax(float v, float* sbuf, int* sibuf) {
  const int lane = threadIdx.x & 31, w = threadIdx.x >> 5;
  int idx = threadIdx.x;
#pragma unroll
  for (int o = 16; o; o >>= 1) {
    const float ov = __shfl_xor(v, o, 32);
    const int oi = __shfl_xor(idx, o, 32);
    if (ov > v || (ov == v && oi < idx)) { v = ov; idx = oi; }
  }
  if (lane == 0) { sbuf[w] = v; sibuf[w] = idx; }
  __syncthreads();
  if (w == 0) {
    float y = (lane < 8) ? sbuf[lane] : -3.0e38f;
    int yi = (lane < 8) ? sibuf[lane] : 0x7fffffff;
#pragma unroll
    for (int o = 4; o; o >>= 1) {
      const float ov = __shfl_xor(y, o, 32);
      const int oi = __shfl_xor(yi, o, 32);
      if (ov > y || (ov == y && oi < yi)) { y = ov; yi = oi; }
    }
    if (lane == 0) sibuf[0] = yi;
  }
  __syncthreads();
  const int r = sibuf[0];
  __syncthreads();
  return r;
}

__global__ __launch_bounds__(256) void softmax_kl(
    const float* __restrict__ logits_s, const float* __restrict__ logits_e,
    const float* __restrict__ b2s, const float* __restrict__ b2e,
    const float* __restrict__ gts, const float* __restrict__ gte,
    const int* __restrict__ vlen, float* __restrict__ out) {
  const int b = blockIdx.x, t = threadIdx.x;
  __shared__ float sbuf[8];
  __shared__ int sibuf[8];
  __shared__ float sp[256];
  const int len = vlen[b];
  const float m = (t < len) ? 1.f : 0.f;
  float il_total = 0.f, aux = 0.f;
  for (int head = 0; head < 2; ++head) {
    const float* lg = head ? logits_e : logits_s;
    const float* gt = head ? gte : gts;
    const float b2 = head ? b2e[0] : b2s[0];
    float* pred = out + 129 + head * 32768 + b * 256;
    const float z = (lg[b * 256 + t] + b2) * m;     // softmax(v*m)
    const float mx = block_max(z, sbuf);
    const float e = expf(z - mx);
    const float s1 = block_sum(e, sbuf);
    const float r = e / s1 * m;
    const float s2 = block_sum(r, sbuf);
    const float p = r / (s2 + 1e-13f) + 1e-13f;
    pred[t] = p;
    sp[t] = p;
    const float g = gt[b * 256 + t];
    const float gq = (m > 0.f) ? g : 1.f;
    const float ret = (m > 0.f) ? gq * logf(p / gq) : 0.f;
    const float il = -block_sum(ret, sbuf);
    const int idx = block_argmax(g, sbuf, sibuf);   // first-occurrence argmax of gt
    const float paux = sp[idx];
    il_total += il;
    aux += logf(paux);
    __syncthreads();
  }
  if (t == 0) {
    out[1 + b] = il_total;           // individual_loss
    atomicAdd(&out[0], -aux);        // total_loss = -sum_b (log ps + log pe)
  }
}

// ---------------------------------------------------------------------------
extern "C" void kernel_launch(void* const* d_in, const int* in_sizes, int n_in,
                              void* d_out, int out_size, void* d_ws, size_t ws_size,
                              hipStream_t stream) {
  (void)in_sizes; (void)n_in; (void)out_size; (void)ws_size;
  const float* tokens    = (const float*)d_in[0];   // [128,20,300]
  const int*   tok_len   = (const int*)d_in[1];     // [128]
  const float* videoFeat = (const float*)d_in[2];   // [128,256,1024]
  const int*   vid_len   = (const int*)d_in[3];     // [128]
  const float* gt_start  = (const float*)d_in[4];   // [128,256]
  const float* gt_end    = (const float*)d_in[5];   // [128,256]
  const float* Wih_l = (const float*)d_in[6];       // [1024,300]
  const float* Whh_l = (const float*)d_in[7];       // [1024,256]
  const float* bih_l = (const float*)d_in[8];
  const float* bhh_l = (const float*)d_in[9];
  const float* Wih_v = (const float*)d_in[10];      // [1024,1024]
  const float* Whh_v = (const float*)d_in[11];      // [1024,256]
  const float* bih_v = (const float*)d_in[12];
  const float* bhh_v = (const float*)d_in[13];
  // d_in[14..17] = loc LSTM params: provably unused by outputs
  const float* W1s = (const float*)d_in[18];        // [256,1024]
  const float* b1s = (const float*)d_in[19];
  const float* W2s = (const float*)d_in[20];        // [1,256]
  const float* b2s = (const float*)d_in[21];
  const float* W1e = (const float*)d_in[22];
  const float* b1e = (const float*)d_in[23];
  const float* W2e = (const float*)d_in[24];
  const float* b2e = (const float*)d_in[25];

  float* ws = (float*)d_ws;
  size_t off = 0;
  float* gates_lang = ws + off; off += (size_t)2560 * 1024;
  float* gates_vid  = ws + off; off += (size_t)32768 * 1024;
  float* Wt2_l  = ws + off; off += (size_t)256 * 1024;
  float* Wt2_v  = ws + off; off += (size_t)256 * 1024;
  float* bsum_l = ws + off; off += 1024;
  float* bsum_v = ws + off; off += 1024;
  float* lang_last = ws + off; off += (size_t)128 * 256;
  float* out_video = ws + off; off += (size_t)32768 * 256;
  float* W1sa_s = ws + off; off += 65536;
  float* W1sb_s = ws + off; off += 65536;
  float* W1sa_e = ws + off; off += 65536;
  float* W1sb_e = ws + off; off += 65536;
  float* P_s = ws + off; off += 32768;
  float* P_e = ws + off; off += 32768;
  float* logits_s = ws + off; off += 32768;
  float* logits_e = ws + off; off += 32768;

  float* out = (float*)d_out;
  // zero atomic targets + "zeros" output section (+ trailing scalar 0)
  hipMemsetAsync(out, 0, sizeof(float), stream);
  hipMemsetAsync(out + 65665, 0, (size_t)32769 * sizeof(float), stream);
  hipMemsetAsync(logits_s, 0, (size_t)2 * 32768 * sizeof(float), stream);

  // prep: transpose+pack Whh, bias sums, head weight folds
  k_transpose_pack<<<1024, 256, 0, stream>>>(Whh_l, Wt2_l);
  k_transpose_pack<<<1024, 256, 0, stream>>>(Whh_v, Wt2_v);
  k_add<<<4, 256, 0, stream>>>(bih_l, bhh_l, bsum_l, 1024);
  k_add<<<4, 256, 0, stream>>>(bih_v, bhh_v, bsum_v, 1024);
  k_w1sum<<<256, 256, 0, stream>>>(W1s, W1sa_s, W1sb_s);
  k_w1sum<<<256, 256, 0, stream>>>(W1e, W1sa_e, W1sb_e);

  // hoisted input-projection GEMMs (WMMA f32)
  gemm_wmma<0><<<dim3(8, 40), 256, 0, stream>>>(
      tokens, Wih_l, bsum_l, gates_lang, nullptr, nullptr, nullptr, 2560, 1024, 300);
  gemm_wmma<0><<<dim3(8, 512), 256, 0, stream>>>(
      videoFeat, Wih_v, bsum_v, gates_vid, nullptr, nullptr, nullptr, 32768, 1024, 1024);

  // recurrences (batch-parallel across 8 workgroups, 16 waves each)
  lstm_rec<<<8, 512, 0, stream>>>(gates_lang, Wt2_l, tok_len, nullptr, lang_last, 20, 0);
  lstm_rec<<<8, 512, 0, stream>>>(gates_vid, Wt2_v, vid_len, out_video, nullptr, 256, 1);

  // per-batch lang contribution to head hidden: P = lang_last @ (W1a+W1c)^T + b1
  gemm_wmma<0><<<dim3(2, 2), 256, 0, stream>>>(
      lang_last, W1sa_s, b1s, P_s, nullptr, nullptr, nullptr, 128, 256, 256);
  gemm_wmma<0><<<dim3(2, 2), 256, 0, stream>>>(
      lang_last, W1sa_e, b1e, P_e, nullptr, nullptr, nullptr, 128, 256, 256);

  // fused head: tanh(out_video@(W1b+W1d)^T + P) . w2 -> logits (atomic)
  gemm_wmma<2><<<dim3(2, 512), 256, 0, stream>>>(
      out_video, W1sb_s, nullptr, nullptr, P_s, W2s, logits_s, 32768, 256, 256);
  gemm_wmma<2><<<dim3(2, 512), 256, 0, stream>>>(
      out_video, W1sb_e, nullptr, nullptr, P_e, W2e, logits_e, 32768, 256, 256);

  // masked softmax + KL + total loss
  softmax_kl<<<128, 256, 0, stream>>>(logits_s, logits_e, b2s, b2e,
                                      gt_start, gt_end, vid_len, out);
}